// EdgeClassification_57312043598044
// MI455X (gfx1250) — compile-verified
//
#include <hip/hip_runtime.h>
#include <hip/hip_bf16.h>

// ---------------------------------------------------------------------------
// EdgeClassification, algebraically refactored:
//   logits[e] = Psrc[src[e]] + Pdst[dst[e]] + c
// where Psrc = emb @ (W_fuse_src @ W_cls), Pdst = emb @ (W_fuse_dst @ W_cls),
//       c    = b_fuse @ W_cls + b_cls
// Node projection GEMM uses V_WMMA_F32_16X16X4_F32 (fp32-exact matrix path).
// Edge kernel is the memory-bound hot loop (~160 MB HBM traffic); index reads
// and output stores are non-temporal so the gather tables stay L2-resident.
// ---------------------------------------------------------------------------

#define N_NODES   50000
#define N_EDGES   1600000
#define DIM       128
#define N_CLASSES 21
#define NPAD      32                 // classes padded to 32 columns
#define N_TILES   (N_NODES / 16)     // 3125 exact
#define PROJ_WAVES 4                 // waves (row tiles) per block in projection

// LDS B layout: [matrix][kpair][col*2 + k&1], kpair row stride padded to 96
// floats so lanes 0-15 (row r) and 16-31 (row r+1) hit disjoint bank halves.
#define KP_STRIDE 96                           // floats per k-pair row
#define MAT_STRIDE ((DIM / 2) * KP_STRIDE)     // 6144 floats per matrix

typedef __attribute__((ext_vector_type(2))) float v2f;
typedef __attribute__((ext_vector_type(8))) float v8f;

// ---------------------------------------------------------------------------
// Kernel 1: fold the two [128,128] fuse weights through W_cls [128,21]
//   Wp[m][k][n] = sum_i Wfuse_m[k][i] * Wcls[i][n]   (n >= 21 zero-padded)
//   c[n]        = sum_i b_fuse[i]     * Wcls[i][n] + b_cls[n]
// ---------------------------------------------------------------------------
__global__ void ec_combine_weights(const float* __restrict__ Wsrc,
                                   const float* __restrict__ Wdst,
                                   const float* __restrict__ Wcls,
                                   const float* __restrict__ bfuse,
                                   const float* __restrict__ bcls,
                                   float* __restrict__ Wp,   // [2*128*32]
                                   float* __restrict__ cvec) // [32]
{
    int t = blockIdx.x * blockDim.x + threadIdx.x;
    if (t < 2 * DIM * NPAD) {
        int n = t & (NPAD - 1);
        int k = (t >> 5) & (DIM - 1);
        int m = t >> 12;
        float v = 0.0f;
        if (n < N_CLASSES) {
            const float* Wf = m ? Wdst : Wsrc;
            #pragma unroll 8
            for (int i = 0; i < DIM; ++i)
                v = fmaf(Wf[k * DIM + i], Wcls[i * N_CLASSES + n], v);
        }
        Wp[t] = v;
    }
    if (t < NPAD) {
        float v = 0.0f;
        if (t < N_CLASSES) {
            #pragma unroll 8
            for (int i = 0; i < DIM; ++i)
                v = fmaf(bfuse[i], Wcls[i * N_CLASSES + t], v);
            v += bcls[t];
        }
        cvec[t] = v;
    }
}

// ---------------------------------------------------------------------------
// Kernel 2: node projection GEMM with V_WMMA_F32_16X16X4_F32.
//   Psrc[N_NODES,32] = emb[N_NODES,128] @ Wp[0]   (Wp[m] is 128x32)
//   Pdst[N_NODES,32] = emb[N_NODES,128] @ Wp[1]
// One wave per 16-row tile; 4 waves/block. Weights are staged into LDS
// pre-shuffled into the WMMA B register layout so each B operand is one
// bank-conflict-free ds_load_b64.
//
// ISA lane layouts (CDNA5 ISA, 32-bit operands):
//   A 16x4 : lanes 0-15 -> M=lane, VGPR0/1 = K0/K1; lanes 16-31 -> K2/K3
//   B 4x16 : lanes 0-15 -> N=lane, VGPR0/1 = K0/K1; lanes 16-31 -> K2/K3
//   C 16x16: VGPR r -> M = r + (lane<16 ? 0 : 8), N = lane & 15
// ---------------------------------------------------------------------------
__global__ __launch_bounds__(32 * PROJ_WAVES)
void ec_node_project(const float* __restrict__ emb,   // [N_NODES,128]
                     const float* __restrict__ Wp,    // [2][128][32]
                     float* __restrict__ Psrc,        // [N_NODES,32]
                     float* __restrict__ Pdst)        // [N_NODES,32]
{
    __shared__ float sW[2 * MAT_STRIDE];   // 48 KB of the 320 KB WGP LDS

    // Stage + shuffle: Wp[m][k][n] -> sW[m*MAT + (k>>1)*KP + n*2 + (k&1)]
    for (int i = threadIdx.x; i < 2 * DIM * NPAD; i += blockDim.x) {
        const int n = i & (NPAD - 1);
        const int k = (i >> 5) & (DIM - 1);
        const int m = i >> 12;
        sW[m * MAT_STRIDE + (k >> 1) * KP_STRIDE + n * 2 + (k & 1)] = Wp[i];
    }
    __syncthreads();

    const int wave = threadIdx.x >> 5;
    const int lane = threadIdx.x & 31;
    const int tile = blockIdx.x * PROJ_WAVES + wave;

    if (tile < N_TILES) {              // wave-uniform: EXEC all-1s inside
        const int m15   = lane & 15;
        const int khi   = lane >> 4;                 // 0 or 1 (k-pair select)
        const int khalf = khi << 1;                  // 0 or 2 (A k offset)
        const float* arow = emb + (tile * 16 + m15) * DIM + khalf;

        v8f acc00 = {}, acc01 = {}, acc10 = {}, acc11 = {};

        #pragma unroll 4
        for (int kb = 0; kb < DIM; kb += 4) {
            v2f a = *(const v2f*)(arow + kb);        // A[M=m15][kk], [kk+1]

            const float* bs = sW + ((kb >> 1) + khi) * KP_STRIDE + m15 * 2;
            const float* bd = bs + MAT_STRIDE;
            v2f b00 = *(const v2f*)(bs);             // src weights, cols 0-15
            v2f b01 = *(const v2f*)(bs + 32);        // src weights, cols 16-31
            v2f b10 = *(const v2f*)(bd);             // dst weights, cols 0-15
            v2f b11 = *(const v2f*)(bd + 32);        // dst weights, cols 16-31

            acc00 = __builtin_amdgcn_wmma_f32_16x16x4_f32(false, a, false, b00,
                                                          (short)0, acc00, false, false);
            acc01 = __builtin_amdgcn_wmma_f32_16x16x4_f32(false, a, false, b01,
                                                          (short)0, acc01, false, false);
            acc10 = __builtin_amdgcn_wmma_f32_16x16x4_f32(false, a, false, b10,
                                                          (short)0, acc10, false, false);
            acc11 = __builtin_amdgcn_wmma_f32_16x16x4_f32(false, a, false, b11,
                                                          (short)0, acc11, false, false);
        }

        const int rbase = tile * 16 + (khi << 3);    // +0 or +8
        #pragma unroll
        for (int i = 0; i < 8; ++i) {
            const int r = rbase + i;
            Psrc[r * NPAD + m15]      = acc00[i];
            Psrc[r * NPAD + 16 + m15] = acc01[i];
            Pdst[r * NPAD + m15]      = acc10[i];
            Pdst[r * NPAD + 16 + m15] = acc11[i];
        }
    }
}

// ---------------------------------------------------------------------------
// Kernel 3: per-edge gather-add. One thread per (edge, class) output element.
// The 134 MB output stream and 12.8 MB index stream are non-temporal so the
// 12.8 MB Psrc/Pdst gather tables stay resident in the 192 MB L2.
// ---------------------------------------------------------------------------
__global__ __launch_bounds__(256)
void ec_edge_logits(const int* __restrict__ ei,       // [2, N_EDGES]
                    const float* __restrict__ Psrc,   // [N_NODES,32]
                    const float* __restrict__ Pdst,   // [N_NODES,32]
                    const float* __restrict__ cvec,   // [32]
                    float* __restrict__ out)          // [N_EDGES, 21]
{
    const int t = blockIdx.x * blockDim.x + threadIdx.x;
    if (t >= N_EDGES * N_CLASSES) return;
    const int e = t / N_CLASSES;           // magic-multiply, no real divide
    const int j = t - e * N_CLASSES;
    const int s = __builtin_nontemporal_load(ei + e);
    const int d = __builtin_nontemporal_load(ei + N_EDGES + e);
    const float v = Psrc[s * NPAD + j] + Pdst[d * NPAD + j] + cvec[j];
    __builtin_nontemporal_store(v, out + t);
}

// ---------------------------------------------------------------------------
extern "C" void kernel_launch(void* const* d_in, const int* in_sizes, int n_in,
                              void* d_out, int out_size, void* d_ws, size_t ws_size,
                              hipStream_t stream) {
    const float* emb   = (const float*)d_in[0];   // [50000,128]
    const int*   eidx  = (const int*)  d_in[1];   // [2,1600000]
    const float* Wsrc  = (const float*)d_in[2];   // [128,128]
    const float* Wdst  = (const float*)d_in[3];   // [128,128]
    const float* bfuse = (const float*)d_in[4];   // [128]
    const float* Wcls  = (const float*)d_in[5];   // [128,21]
    const float* bcls  = (const float*)d_in[6];   // [21]
    float* out = (float*)d_out;                   // [1600000,21]

    // Workspace layout (~12.83 MB)
    float* Wp   = (float*)d_ws;                   // 2*128*32
    float* cvec = Wp + 2 * DIM * NPAD;            // 32
    float* Psrc = cvec + NPAD;                    // 50000*32
    float* Pdst = Psrc + N_NODES * NPAD;          // 50000*32

    // 1) Fold weights through the classifier.
    ec_combine_weights<<<(2 * DIM * NPAD + 255) / 256, 256, 0, stream>>>(
        Wsrc, Wdst, Wcls, bfuse, bcls, Wp, cvec);

    // 2) Per-node projections via fp32 WMMA. 3125 tiles / 4 waves per block.
    ec_node_project<<<(N_TILES + PROJ_WAVES - 1) / PROJ_WAVES, 32 * PROJ_WAVES,
                      0, stream>>>(emb, Wp, Psrc, Pdst);

    // 3) Per-edge gather-add (memory-bound, coalesced NT output stream).
    const int total = N_EDGES * N_CLASSES;
    ec_edge_logits<<<(total + 255) / 256, 256, 0, stream>>>(
        eidx, Psrc, Pdst, cvec, out);
}